// Transformer_72095321030981
// MI455X (gfx1250) — compile-verified
//
#include <hip/hip_runtime.h>
#include <hip/hip_bf16.h>
#include <cstdint>
#include <cstddef>

// ---------------- problem constants ----------------
#define DD   512        // model dim
#define HH   8          // heads
#define DKK  64         // head dim
#define FF   2048       // ffn dim
#define NL   4          // layers
#define VV   32000      // vocab
#define BB   4          // batch
#define SS   1024       // src len == trg_input len
#define NTOK (BB*SS)    // 4096 tokens per side
#define KT   16         // GEMM k-chunk staged through LDS

typedef float v2f __attribute__((ext_vector_type(2)));
typedef float v8f __attribute__((ext_vector_type(8)));

// D = A(16x4) * B(4x16) + C, full fp32 on the matrix pipe.
// A-frag: m=lane%16, k=2*(lane/16)+r ; B-frag: n=lane%16, k=2*(lane/16)+r
// C/D:    m=j+8*(lane/16), n=lane%16
__device__ __forceinline__ v8f wmma_f32(v2f a, v2f b, v8f c) {
  return __builtin_amdgcn_wmma_f32_16x16x4_f32(false, a, false, b, (short)0, c,
                                               false, false);
}

// CDNA5 async global->LDS copy (ASYNCcnt). 16B per lane.
__device__ __forceinline__ void async_copy_b128(void* lds, const float* g) {
  unsigned l = (unsigned)(uintptr_t)lds;   // low 32 bits = LDS offset
  asm volatile("global_load_async_to_lds_b128 %0, %1, off"
               :: "v"(l), "v"(g) : "memory");
}
__device__ __forceinline__ void wait_async_le1() {
  asm volatile("s_wait_asynccnt 0x1" ::: "memory");
}
__device__ __forceinline__ void wait_async_0() {
  asm volatile("s_wait_asynccnt 0x0" ::: "memory");
}

// ---------------- embedding + positional encoding ----------------
__global__ void embed_kernel(const int* __restrict__ tok, const float* __restrict__ emb,
                             const float* __restrict__ pe, float* __restrict__ out,
                             int seq_stride) {
  int t = blockIdx.x;                 // 0..4095
  int b = t >> 10, s = t & 1023;
  int id = tok[b * seq_stride + s];
  const float* e = emb + (size_t)id * DD;
  const float* p = pe + (size_t)s * DD;
  float* o = out + (size_t)t * DD;
  for (int d = threadIdx.x; d < DD; d += blockDim.x) o[d] = e[d] + p[d];
}

__global__ void valid_kernel(const int* __restrict__ tok, float* __restrict__ valid,
                             int seq_stride) {
  int t = blockIdx.x * blockDim.x + threadIdx.x;
  if (t < NTOK) {
    int b = t >> 10, s = t & 1023;
    valid[t] = (tok[b * seq_stride + s] != 0) ? 1.0f : 0.0f;
  }
}

// ---------------- layernorm: one block (256 thr) per token ----------------
__global__ void ln_kernel(const float* __restrict__ x, const float* __restrict__ w,
                          const float* __restrict__ bias, float* __restrict__ y) {
  __shared__ float red[256];
  int t = blockIdx.x;
  const float* xi = x + (size_t)t * DD;
  float v0 = xi[threadIdx.x], v1 = xi[threadIdx.x + 256];
  red[threadIdx.x] = v0 + v1;
  __syncthreads();
  for (int o = 128; o > 0; o >>= 1) {
    if (threadIdx.x < o) red[threadIdx.x] += red[threadIdx.x + o];
    __syncthreads();
  }
  float mean = red[0] * (1.0f / DD);
  __syncthreads();
  float d0 = v0 - mean, d1 = v1 - mean;
  red[threadIdx.x] = d0 * d0 + d1 * d1;
  __syncthreads();
  for (int o = 128; o > 0; o >>= 1) {
    if (threadIdx.x < o) red[threadIdx.x] += red[threadIdx.x + o];
    __syncthreads();
  }
  float r = rsqrtf(red[0] * (1.0f / DD) + 1e-5f);
  float* yo = y + (size_t)t * DD;
  yo[threadIdx.x]       = d0 * r * w[threadIdx.x]       + bias[threadIdx.x];
  yo[threadIdx.x + 256] = d1 * r * w[threadIdx.x + 256] + bias[threadIdx.x + 256];
}

__global__ void add_kernel(const float* __restrict__ a, const float* __restrict__ b,
                           float* __restrict__ o, int n) {
  int i = blockIdx.x * blockDim.x + threadIdx.x;
  if (i < n) o[i] = a[i] + b[i];
}

// ---------------- fp32 WMMA GEMM with async-LDS staged B ----------------
// C[M,N] = A[M,K] @ W[K,N] + bias.  Block = 8 waves -> 128 rows x 64 cols.
// All 8 waves share one 16x64 B-chunk, double-buffered in LDS via
// GLOBAL_LOAD_ASYNC_TO_LDS_B128 (one b128 per thread per chunk).
// Requires M % 128 == 0, N % 64 == 0, K % 16 == 0 (true for all launches).
template <int RELU>
__global__ __launch_bounds__(256) void gemm_kernel(
    const float* __restrict__ A, const float* __restrict__ W,
    const float* __restrict__ bias, float* __restrict__ C, int M, int N, int K) {
  __shared__ float Bt[2][KT * 64];   // 2 x 4 KB
  int lane = threadIdx.x;            // 0..31
  int wave = threadIdx.y;            // 0..7
  int tid = wave * 32 + lane;
  int m0 = (blockIdx.y * 8 + wave) * 16;
  int n0 = blockIdx.x * 64;
  int lm = lane & 15, lg = lane >> 4;

  // cooperative B loader: thread -> (row lr of chunk, 4-float segment)
  int lr  = tid >> 4;                // 0..15
  int lcb = (tid & 15) << 2;         // 0,4,...,60
  const float* wbase = W + (size_t)lr * N + n0 + lcb;

  v8f acc[4] = {};
  const float* arow = A + (size_t)(m0 + lm) * K + 2 * lg;

  int nchunks = K / KT;
  async_copy_b128(&Bt[0][lr * 64 + lcb], wbase);           // prefetch chunk 0
  for (int c = 0; c < nchunks; ++c) {
    int bi = c & 1;
    if (c + 1 < nchunks) {
      async_copy_b128(&Bt[bi ^ 1][lr * 64 + lcb],
                      wbase + (size_t)(c + 1) * KT * N);   // prefetch next
      wait_async_le1();   // in-order completion => chunk c has landed
    } else {
      wait_async_0();
    }
    __syncthreads();      // whole block's chunk-c segments visible
    const float* bl = &Bt[bi][0];
    int k0 = c * KT;
#pragma unroll
    for (int ks = 0; ks < KT; ks += 4) {
      v2f a = *(const v2f*)(arow + k0 + ks);
      const float* brow = bl + (ks + 2 * lg) * 64 + lm;
#pragma unroll
      for (int t = 0; t < 4; t++) {
        v2f b;
        b.x = brow[16 * t];
        b.y = brow[16 * t + 64];
        acc[t] = wmma_f32(a, b, acc[t]);
      }
    }
    __syncthreads();      // all reads of buf bi done before c+2 overwrites it
  }

#pragma unroll
  for (int t = 0; t < 4; t++) {
    int n = n0 + 16 * t + lm;
    float bv = bias[n];
#pragma unroll
    for (int j = 0; j < 8; j++) {
      int m = m0 + j + 8 * lg;
      float v = acc[t][j] + bv;
      if (RELU) v = fmaxf(v, 0.0f);
      C[(size_t)m * N + n] = v;
    }
  }
}

// ---------------- fused flash attention (fp32 WMMA) ----------------
// One wave = one 16-query tile of one (b,h). Streams 16-key tiles:
//   S = Q*K^T (16 wmma), online softmax (16-lane shuffle reductions),
//   P transposed via wave-local LDS, O += P*V (16 wmma).
__global__ __launch_bounds__(128) void attn_kernel(
    const float* __restrict__ Q, const float* __restrict__ Kb,
    const float* __restrict__ Vb, float* __restrict__ O,
    const float* __restrict__ kvalid, int causal) {
  __shared__ float psh[4 * 256];
  int lane = threadIdx.x, wave = threadIdx.y;
  int lm = lane & 15, lg = lane >> 4;
  int sq0 = (blockIdx.x * 4 + wave) * 16;
  int h = blockIdx.y, b = blockIdx.z;
  size_t hd = (size_t)h * DKK;
  float* pl = psh + wave * 256;

  // preload Q tile (16x64) as 16 A-frags
  float qa[16][2];
  {
    const float* qp = Q + ((size_t)(b * SS + sq0 + lm)) * DD + hd + 2 * lg;
#pragma unroll
    for (int kc = 0; kc < 16; kc++) {
      v2f t = *(const v2f*)(qp + 4 * kc);
      qa[kc][0] = t.x; qa[kc][1] = t.y;
    }
  }

  float mrow[8], lrow[8];
#pragma unroll
  for (int j = 0; j < 8; j++) { mrow[j] = -1e30f; lrow[j] = 0.0f; }
  v8f o[4] = {};

  int ktmax = causal ? (sq0 / 16) : (SS / 16 - 1);
  for (int kt = 0; kt <= ktmax; kt++) {
    int sk0 = kt * 16;
    // scores = Q * K^T  (B element (k,n) = K[sk0+n, hd+k])
    v8f sc = {};
    {
      const float* kp = Kb + ((size_t)(b * SS + sk0 + lm)) * DD + hd + 2 * lg;
#pragma unroll
      for (int kc = 0; kc < 16; kc++) {
        v2f a; a.x = qa[kc][0]; a.y = qa[kc][1];
        v2f bb = *(const v2f*)(kp + 4 * kc);
        sc = wmma_f32(a, bb, sc);
      }
    }
    // scale, mask, online softmax (row m lives across one 16-lane group)
    float kv = kvalid[b * SS + sk0 + lm];
    float corr[8], pj[8];
#pragma unroll
    for (int j = 0; j < 8; j++) {
      float s = sc[j] * 0.125f;                       // 1/sqrt(64)
      int m = sq0 + j + 8 * lg;
      int skn = sk0 + lm;
      bool ok = (kv != 0.0f) && (!causal || (skn <= m));
      s = ok ? s : -1e30f;
      float rm = s;
      rm = fmaxf(rm, __shfl_xor(rm, 1, 32));
      rm = fmaxf(rm, __shfl_xor(rm, 2, 32));
      rm = fmaxf(rm, __shfl_xor(rm, 4, 32));
      rm = fmaxf(rm, __shfl_xor(rm, 8, 32));
      float mnew = fmaxf(mrow[j], rm);
      float e = __expf(s - mnew);
      float rs = e;
      rs += __shfl_xor(rs, 1, 32);
      rs += __shfl_xor(rs, 2, 32);
      rs += __shfl_xor(rs, 4, 32);
      rs += __shfl_xor(rs, 8, 32);
      float c = __expf(mrow[j] - mnew);
      lrow[j] = lrow[j] * c + rs;
      mrow[j] = mnew;
      corr[j] = c;
      pj[j] = e;
    }
#pragma unroll
    for (int t = 0; t < 4; t++)
#pragma unroll
      for (int j = 0; j < 8; j++) o[t][j] *= corr[j];

    // wave-local transpose of P through LDS (C-layout -> A-layout)
#pragma unroll
    for (int j = 0; j < 8; j++) pl[(j + 8 * lg) * 16 + lm] = pj[j];
    __builtin_amdgcn_wave_barrier();        // LDS is in-order within a wave

    // O += P * V   (B element (k,n) = V[sk0+k, hd+16t+n])
    const float* vp = Vb + ((size_t)(b * SS + sk0)) * DD + hd;
#pragma unroll
    for (int kc = 0; kc < 4; kc++) {
      int kk = 4 * kc + 2 * lg;
      v2f a = *(const v2f*)(pl + lm * 16 + kk);
      const float* vr = vp + (size_t)kk * DD + lm;
#pragma unroll
      for (int t = 0; t < 4; t++) {
        v2f bb;
        bb.x = vr[16 * t];
        bb.y = vr[16 * t + DD];
        o[t] = wmma_f32(a, bb, o[t]);
      }
    }
    __builtin_amdgcn_wave_barrier();        // WAR guard before next tile's P store
  }

  float inv[8];
#pragma unroll
  for (int j = 0; j < 8; j++) inv[j] = 1.0f / lrow[j];
  float* op = O + ((size_t)(b * SS)) * DD + hd;
#pragma unroll
  for (int t = 0; t < 4; t++)
#pragma unroll
    for (int j = 0; j < 8; j++)
      op[(size_t)(sq0 + j + 8 * lg) * DD + 16 * t + lm] = o[t][j] * inv[j];
}

// ---------------- loss: per-token logsumexp NLL, deterministic reduce ----------------
__global__ void nll_kernel(const float* __restrict__ logits, const int* __restrict__ trg,
                           float* __restrict__ nll) {
  __shared__ float red[256];
  int t = blockIdx.x;
  int b = t >> 10, s = t & 1023;
  int label = trg[b * 1025 + s + 1];       // trg_label = trg[:, 1:]
  const float* lp = logits + (size_t)t * VV;
  float mx = -1e30f;
  for (int i = threadIdx.x; i < VV; i += 256) mx = fmaxf(mx, lp[i]);
  red[threadIdx.x] = mx;
  __syncthreads();
  for (int o = 128; o > 0; o >>= 1) {
    if (threadIdx.x < o) red[threadIdx.x] = fmaxf(red[threadIdx.x], red[threadIdx.x + o]);
    __syncthreads();
  }
  mx = red[0];
  __syncthreads();
  float sum = 0.0f;
  for (int i = threadIdx.x; i < VV; i += 256) sum += __expf(lp[i] - mx);
  red[threadIdx.x] = sum;
  __syncthreads();
  for (int o = 128; o > 0; o >>= 1) {
    if (threadIdx.x < o) red[threadIdx.x] += red[threadIdx.x + o];
    __syncthreads();
  }
  if (threadIdx.x == 0) nll[t] = (__logf(red[0]) + mx) - lp[label];
}

__global__ void loss_kernel(const float* __restrict__ nll, float* __restrict__ out) {
  __shared__ float red[256];
  float s = 0.0f;
  for (int i = threadIdx.x; i < NTOK; i += 256) s += nll[i];
  red[threadIdx.x] = s;
  __syncthreads();
  for (int o = 128; o > 0; o >>= 1) {
    if (threadIdx.x < o) red[threadIdx.x] += red[threadIdx.x + o];
    __syncthreads();
  }
  if (threadIdx.x == 0) out[0] = red[0] * (1.0f / NTOK);
}

// ---------------- host orchestration ----------------
struct LayerP {
  const float *qw, *qb, *kw, *kb, *vw, *vb, *ow, *ob;
  const float *l1w, *l1b, *l2w, *l2b, *ln1w, *ln1b, *ln2w, *ln2b;
};

extern "C" void kernel_launch(void* const* d_in, const int* in_sizes, int n_in,
                              void* d_out, int out_size, void* d_ws, size_t ws_size,
                              hipStream_t stream) {
  (void)n_in; (void)out_size; (void)ws_size;

  // -------- input-layout detection (pytree flatten order is ambiguous) --------
  const int* src; const int* trg; const float* pe;
  int pbase;
  if (in_sizes[0] == NTOK) {             // top-level insertion order: src,trg,params,pe
    src = (const int*)d_in[0];
    trg = (const int*)d_in[1];
    pbase = 2;
    pe = (const float*)d_in[2 + 132];
  } else {                               // fully sorted: params(132), pe, src, trg
    pbase = 0;
    pe  = (const float*)d_in[132];
    src = (const int*)d_in[133];
    trg = (const int*)d_in[134];
  }
  bool sorted_leaves = (in_sizes[pbase] == DD);  // sorted => first leaf is dec0.attn.k.b (512)

  const float *emb_in, *emb_out, *fcw, *fcb;
  LayerP encp[NL], decp[NL];
  if (sorted_leaves) {
    // jax sorted: dec(4x16), emb_in, emb_out, enc(4x16), fc.b, fc.w
    // per-layer: k.b,k.w,o.b,o.w,q.b,q.w,v.b,v.w,l1.b,l1.w,l2.b,l2.w,ln1.b,ln1.w,ln2.b,ln2.w
    auto fill = [&](LayerP* Ls, int base) {
      for (int l = 0; l < NL; l++) {
        void* const* p = d_in + base + l * 16;
        Ls[l] = { (const float*)p[5], (const float*)p[4], (const float*)p[1], (const float*)p[0],
                  (const float*)p[7], (const float*)p[6], (const float*)p[3], (const float*)p[2],
                  (const float*)p[9], (const float*)p[8], (const float*)p[11], (const float*)p[10],
                  (const float*)p[13], (const float*)p[12], (const float*)p[15], (const float*)p[14] };
      }
    };
    fill(decp, pbase);
    emb_in  = (const float*)d_in[pbase + 64];
    emb_out = (const float*)d_in[pbase + 65];
    fill(encp, pbase + 66);
    fcb = (const float*)d_in[pbase + 130];
    fcw = (const float*)d_in[pbase + 131];
  } else {
    // insertion order: emb_in, emb_out, enc(4x16), dec(4x16), fc.w, fc.b
    // per-layer: q.w,q.b,k.w,k.b,v.w,v.b,o.w,o.b,l1.w,l1.b,l2.w,l2.b,ln1.w,ln1.b,ln2.w,ln2.b
    emb_in  = (const float*)d_in[pbase];
    emb_out = (const float*)d_in[pbase + 1];
    auto fill = [&](LayerP* Ls, int base) {
      for (int l = 0; l < NL; l++) {
        void* const* p = d_in + base + l * 16;
        Ls[l] = { (const float*)p[0], (const float*)p[1], (const float*)p[2], (const float*)p[3],
                  (const float*)p[4], (const float*)p[5], (const float*)p[6], (const float*)p[7],
                  (const float*)p[8], (const float*)p[9], (const float*)p[10], (const float*)p[11],
                  (const float*)p[12], (const float*)p[13], (const float*)p[14], (const float*)p[15] };
      }
    };
    fill(encp, pbase + 2);
    fill(decp, pbase + 2 + 64);
    fcw = (const float*)d_in[pbase + 130];
    fcb = (const float*)d_in[pbase + 131];
  }

  // -------- workspace layout (fp32 bump allocator, ~110 MB) --------
  float* wsp = (float*)d_ws;
  size_t off = 0;
  auto alloc = [&](size_t n) { float* r = wsp + off; off += n; return r; };
  float* xenc = alloc((size_t)NTOK * DD);   // encoder state / memory
  float* ydec = alloc((size_t)NTOK * DD);   // decoder state
  float* lnb  = alloc((size_t)NTOK * DD);   // layernorm output
  float* x2a  = alloc((size_t)NTOK * DD);   // x + self-attn
  float* x2   = alloc((size_t)NTOK * DD);   // ln2(x + sa)
  float* qb_  = alloc((size_t)NTOK * DD);
  float* kb_  = alloc((size_t)NTOK * DD);
  float* vb_  = alloc((size_t)NTOK * DD);
  float* ab_  = alloc((size_t)NTOK * DD);   // attention output (pre O-proj)
  float* pj_  = alloc((size_t)NTOK * DD);   // projection / residual temp
  float* ffh  = alloc((size_t)NTOK * FF);   // ffn hidden
  float* vsrc = alloc(NTOK);
  float* vtrg = alloc(NTOK);
  float* nllb = alloc(NTOK);
  float* logits = (float*)d_out;            // [4096, 32000]
  float* lossp  = logits + (size_t)NTOK * VV;

  auto gemm = [&](const float* A, const float* W, const float* bias, float* C,
                  int M, int N, int K, bool relu) {
    dim3 grid(N / 64, M / 128), blk(32, 8);
    if (relu) gemm_kernel<1><<<grid, blk, 0, stream>>>(A, W, bias, C, M, N, K);
    else      gemm_kernel<0><<<grid, blk, 0, stream>>>(A, W, bias, C, M, N, K);
  };
  auto attn = [&](const float* Q, const float* K, const float* Vv, float* O,
                  const float* kvalid, int causal) {
    attn_kernel<<<dim3(SS / 64, HH, BB), dim3(32, 4), 0, stream>>>(Q, K, Vv, O, kvalid, causal);
  };
  auto lnorm = [&](const float* x, const float* w, const float* b, float* y) {
    ln_kernel<<<NTOK, 256, 0, stream>>>(x, w, b, y);
  };
  auto residual = [&](const float* a, const float* b, float* o) {
    int n = NTOK * DD;
    add_kernel<<<(n + 255) / 256, 256, 0, stream>>>(a, b, o, n);
  };

  // -------- embeddings + masks --------
  embed_kernel<<<NTOK, 256, 0, stream>>>(src, emb_in, pe, xenc, 1024);
  embed_kernel<<<NTOK, 256, 0, stream>>>(trg, emb_out, pe, ydec, 1025);  // trg_input = trg[:, :-1]
  valid_kernel<<<(NTOK + 255) / 256, 256, 0, stream>>>(src, vsrc, 1024);
  valid_kernel<<<(NTOK + 255) / 256, 256, 0, stream>>>(trg, vtrg, 1025);

  // -------- encoder --------
  for (int l = 0; l < NL; l++) {
    const LayerP& P = encp[l];
    lnorm(xenc, P.ln1w, P.ln1b, lnb);
    gemm(lnb, P.qw, P.qb, qb_, NTOK, DD, DD, false);
    gemm(lnb, P.kw, P.kb, kb_, NTOK, DD, DD, false);
    gemm(lnb, P.vw, P.vb, vb_, NTOK, DD, DD, false);
    attn(qb_, kb_, vb_, ab_, vsrc, 0);
    gemm(ab_, P.ow, P.ob, pj_, NTOK, DD, DD, false);
    residual(xenc, pj_, xenc);                       // o1 = x + a
    lnorm(xenc, P.ln2w, P.ln2b, lnb);
    gemm(lnb, P.l1w, P.l1b, ffh, NTOK, FF, DD, true);   // relu
    gemm(ffh, P.l2w, P.l2b, pj_, NTOK, DD, FF, false);
    residual(xenc, pj_, xenc);                       // out = ffn + o1
  }

  // -------- decoder --------
  for (int l = 0; l < NL; l++) {
    const LayerP& P = decp[l];
    lnorm(ydec, P.ln1w, P.ln1b, lnb);                // a = ln1(x)
    gemm(lnb, P.qw, P.qb, qb_, NTOK, DD, DD, false);
    gemm(lnb, P.kw, P.kb, kb_, NTOK, DD, DD, false);
    gemm(lnb, P.vw, P.vb, vb_, NTOK, DD, DD, false);
    attn(qb_, kb_, vb_, ab_, vtrg, 1);               // causal self-attn
    gemm(ab_, P.ow, P.ob, pj_, NTOK, DD, DD, false); // sa
    residual(ydec, pj_, x2a);                        // x + sa
    lnorm(x2a, P.ln2w, P.ln2b, x2);                  // x2 = ln2(x + sa)
    // cross attention: q from x2, k/v from encoder memory (raw)
    gemm(x2,   P.qw, P.qb, qb_, NTOK, DD, DD, false);
    gemm(xenc, P.kw, P.kb, kb_, NTOK, DD, DD, false);
    gemm(xenc, P.vw, P.vb, vb_, NTOK, DD, DD, false);
    attn(qb_, kb_, vb_, ab_, vsrc, 0);
    gemm(ab_, P.ow, P.ob, pj_, NTOK, DD, DD, false); // ca
    residual(x2, pj_, ydec);                         // ff_in = x2 + ca
    lnorm(ydec, P.ln2w, P.ln2b, lnb);
    gemm(lnb, P.l1w, P.l1b, ffh, NTOK, FF, DD, true);
    gemm(ffh, P.l2w, P.l2b, pj_, NTOK, DD, FF, false);
    residual(ydec, pj_, ydec);                       // out = ffn + ff_in
  }

  // -------- vocab projection + loss --------
  gemm(ydec, fcw, fcb, logits, NTOK, VV, DD, false); // 4096 x 32000
  nll_kernel<<<NTOK, 256, 0, stream>>>(logits, trg, nllb);
  loss_kernel<<<1, 256, 0, stream>>>(nllb, lossp);
}